// DecoderBlock_63479616634988
// MI455X (gfx1250) — compile-verified
//
#include <hip/hip_runtime.h>

typedef float v2f __attribute__((ext_vector_type(2)));
typedef float v8f __attribute__((ext_vector_type(8)));
typedef int i4v __attribute__((vector_size(16)));  // matches builtin param type

// D = A(16x4 f32) * B(4x16 f32) + C(16x16 f32), f32 matrix-core path.
__device__ __forceinline__ v8f wmma_f32_4(v2f a, v2f b, v8f c) {
  return __builtin_amdgcn_wmma_f32_16x16x4_f32(false, a, false, b, (short)0, c,
                                               false, false);
}

// Async global->LDS staging (CDNA5). Falls back to a VGPR round-trip copy.
#if __has_builtin(__builtin_amdgcn_global_load_async_to_lds_b128) && \
    __has_builtin(__builtin_amdgcn_s_wait_asynccnt)
#define USE_ASYNC_LDS 1
#endif

__device__ __forceinline__ void cp16_g2l(const void* g, void* l) {
#ifdef USE_ASYNC_LDS
  __builtin_amdgcn_global_load_async_to_lds_b128((i4v*)g, (i4v*)l, 0, 0);
#else
  *(float4*)l = *(const float4*)g;
#endif
}
__device__ __forceinline__ void cp_wait() {
#ifdef USE_ASYNC_LDS
  __builtin_amdgcn_s_wait_asynccnt(0);
#endif
}

// ---------------------------------------------------------------------------
// Y[R,64] = act(X[R,64] @ W[64,64] + b)
// ACT: 0 = none, 1 = LayerNorm+ReLU, 2 = ReLU
// 256 threads (8 waves). Each wave: 16 rows x 64 cols (4 WMMA tiles, K=64).
// ---------------------------------------------------------------------------
template <int ACT>
__global__ __launch_bounds__(256) void linear_kernel(
    const float* __restrict__ X, const float* __restrict__ W,
    const float* __restrict__ bias, const float* __restrict__ lnw,
    const float* __restrict__ lnb, float* __restrict__ Y, int R) {
  __shared__ float sX[128][68];  // X tile, later reused for raw Y
  __shared__ float sW[64][68];
  __shared__ float sB[64];
  __shared__ float sLw[64];
  __shared__ float sLb[64];

  const int tid = threadIdx.x;
  const int r0 = blockIdx.x * 128;

  for (int i = tid; i < 1024; i += 256) {  // W: 64x64 = 1024 float4
    int row = i >> 4, c4 = (i & 15) << 2;
    cp16_g2l(W + row * 64 + c4, &sW[row][c4]);
  }
  for (int i = tid; i < 2048; i += 256) {  // X: 128x64 = 2048 float4
    int row = i >> 4, c4 = (i & 15) << 2;
    if (r0 + row < R) {
      cp16_g2l(X + (size_t)(r0 + row) * 64 + c4, &sX[row][c4]);
    } else {
      *(float4*)&sX[row][c4] = make_float4(0.f, 0.f, 0.f, 0.f);
    }
  }
  if (tid < 64) {
    sB[tid] = bias[tid];
    if (ACT == 1) {
      sLw[tid] = lnw[tid];
      sLb[tid] = lnb[tid];
    }
  }
  cp_wait();
  __syncthreads();

  const int wave = __builtin_amdgcn_readfirstlane(tid >> 5);  // uniform
  const int lane = tid & 31;
  const int m16 = lane & 15;
  const int khalf = (lane >> 4) << 1;  // 0 or 2
  const int hi8 = (lane >> 4) << 3;    // 0 or 8
  const int rb = wave << 4;

  // preload all 16 A-fragments for this wave's 16-row strip
  v2f af[16];
#pragma unroll
  for (int kk = 0; kk < 16; ++kk)
    af[kk] = *(const v2f*)&sX[rb + m16][kk * 4 + khalf];

#pragma unroll
  for (int ct = 0; ct < 4; ++ct) {
    v2f bf[16];
#pragma unroll
    for (int kk = 0; kk < 16; ++kk) {
      v2f b;
      b[0] = sW[kk * 4 + khalf][ct * 16 + m16];
      b[1] = sW[kk * 4 + khalf + 1][ct * 16 + m16];
      bf[kk] = b;
    }
    v8f acc = (v8f){0, 0, 0, 0, 0, 0, 0, 0};
#pragma unroll
    for (int kk = 0; kk < 16; ++kk) acc = wmma_f32_4(af[kk], bf[kk], acc);
#pragma unroll
    for (int v = 0; v < 8; ++v) sX[rb + v + hi8][ct * 16 + m16] = acc[v];
  }
  __syncthreads();

  if (ACT == 1) {
    if (tid < 128 && r0 + tid < R) {
      float vb[64];
      const float4* rp = (const float4*)&sX[tid][0];
      float mean = 0.f;
#pragma unroll
      for (int i = 0; i < 16; ++i) {
        float4 t4 = rp[i];
        vb[4 * i + 0] = t4.x + sB[4 * i + 0];
        vb[4 * i + 1] = t4.y + sB[4 * i + 1];
        vb[4 * i + 2] = t4.z + sB[4 * i + 2];
        vb[4 * i + 3] = t4.w + sB[4 * i + 3];
        mean += vb[4 * i + 0] + vb[4 * i + 1] + vb[4 * i + 2] + vb[4 * i + 3];
      }
      mean *= (1.f / 64.f);
      float var = 0.f;
#pragma unroll
      for (int c = 0; c < 64; ++c) {
        float d = vb[c] - mean;
        var += d * d;
      }
      float inv = rsqrtf(var * (1.f / 64.f) + 1e-5f);
      float4* yr = (float4*)(Y + (size_t)(r0 + tid) * 64);
#pragma unroll
      for (int i = 0; i < 16; ++i) {
        float4 o;
        o.x = fmaxf((vb[4 * i + 0] - mean) * inv * sLw[4 * i + 0] + sLb[4 * i + 0], 0.f);
        o.y = fmaxf((vb[4 * i + 1] - mean) * inv * sLw[4 * i + 1] + sLb[4 * i + 1], 0.f);
        o.z = fmaxf((vb[4 * i + 2] - mean) * inv * sLw[4 * i + 2] + sLb[4 * i + 2], 0.f);
        o.w = fmaxf((vb[4 * i + 3] - mean) * inv * sLw[4 * i + 3] + sLb[4 * i + 3], 0.f);
        yr[i] = o;
      }
    }
  } else {
    for (int i = tid; i < 2048; i += 256) {
      int row = i >> 4, c4 = (i & 15) << 2;
      if (r0 + row >= R) continue;
      float4 t4 = *(const float4*)&sX[row][c4];
      float4 o;
      o.x = t4.x + sB[c4 + 0];
      o.y = t4.y + sB[c4 + 1];
      o.z = t4.z + sB[c4 + 2];
      o.w = t4.w + sB[c4 + 3];
      if (ACT == 2) {
        o.x = fmaxf(o.x, 0.f);
        o.y = fmaxf(o.y, 0.f);
        o.z = fmaxf(o.z, 0.f);
        o.w = fmaxf(o.w, 0.f);
      }
      *(float4*)(Y + (size_t)(r0 + row) * 64 + c4) = o;
    }
  }
}

// ---------------------------------------------------------------------------
// Grouped vector attention, one workgroup = QB=4 query points x K=16 pairs.
// ---------------------------------------------------------------------------
__global__ __launch_bounds__(256) void gva_kernel(
    const float* __restrict__ qbuf, const float* __restrict__ kbuf,
    const float* __restrict__ vbuf, const float* __restrict__ xyz,
    const float* __restrict__ newxyz, const int* __restrict__ refidx,
    const float* __restrict__ pb_l1w, const float* __restrict__ pb_l1b,
    const float* __restrict__ pb_lnw, const float* __restrict__ pb_lnb,
    const float* __restrict__ pb_l2w, const float* __restrict__ pb_l2b,
    const float* __restrict__ we_l1w, const float* __restrict__ we_l1b,
    const float* __restrict__ we_lnw, const float* __restrict__ we_lnb,
    const float* __restrict__ we_l2w, const float* __restrict__ we_l2b,
    float* __restrict__ xout) {
  constexpr int QB = 4, KN = 16, PP = QB * KN;  // 64 pair-rows

  __shared__ float t1[PP][68];    // l1-out / rel / (value+peb)
  __shared__ float peb[PP][68];   // positional bias
  __shared__ float4 posb[PP];     // (dx,dy,dz,1)
  __shared__ int idxb[PP];
  __shared__ float qs[QB][64];
  __shared__ float wl[PP][16];    // we.l1 out (padded), then weights
  __shared__ float sW2[64][68];   // pb.l2.w
  __shared__ float sBl1[4][64];   // pb.l1.w rows 0..2, bias row 3
  __shared__ float sWe[64][16];   // we.l1.w zero-padded to 16 cols

  const int tid = threadIdx.x;
  const int n0 = blockIdx.x * QB;
  const int wave = __builtin_amdgcn_readfirstlane(tid >> 5);  // uniform
  const int lane = tid & 31;
  const int m16 = lane & 15;
  const int khalf = (lane >> 4) << 1;
  const int hi8 = (lane >> 4) << 3;

  // --- stage weights / queries (async where possible) ---
  for (int i = tid; i < 1024; i += 256) {
    int row = i >> 4, c4 = (i & 15) << 2;
    cp16_g2l(pb_l2w + row * 64 + c4, &sW2[row][c4]);
  }
  if (tid < 64) {
    int qn = tid >> 4, c4 = (tid & 15) << 2;
    cp16_g2l(qbuf + (size_t)(n0 + qn) * 64 + c4, &qs[qn][c4]);
  }
  if (tid < 192)
    sBl1[tid >> 6][tid & 63] = pb_l1w[tid];
  else
    sBl1[3][tid - 192] = pb_l1b[tid - 192];
  for (int i = tid; i < 1024; i += 256) {
    int k = i >> 4, n = i & 15;
    sWe[k][n] = (n < 8) ? we_l1w[k * 8 + n] : 0.f;
  }
  // pair data: relative position + index; prefetch gathered rows
  if (tid < PP) {
    int qn = tid >> 4, k = tid & 15;
    int gi = n0 + qn;
    int idx = refidx[gi * 16 + k];
    int ic = idx < 0 ? 0 : idx;
    idxb[tid] = idx;
    float px = xyz[(size_t)ic * 3 + 0] - newxyz[(size_t)gi * 3 + 0];
    float py = xyz[(size_t)ic * 3 + 1] - newxyz[(size_t)gi * 3 + 1];
    float pz = xyz[(size_t)ic * 3 + 2] - newxyz[(size_t)gi * 3 + 2];
    posb[tid] = make_float4(px, py, pz, 1.f);
    __builtin_prefetch(kbuf + (size_t)ic * 64, 0, 3);
    __builtin_prefetch(vbuf + (size_t)ic * 64, 0, 3);
  }
  cp_wait();
  __syncthreads();

  // --- pb.l1: [64,4] @ [4,64], bias folded as K row 3; 2 tiles per wave ---
#pragma unroll
  for (int tt = 0; tt < 2; ++tt) {
    int t = wave + tt * 8;
    int rt = t >> 2, ct = t & 3;
    const float* pp = (const float*)&posb[rt * 16 + m16];
    v2f a = *(const v2f*)&pp[khalf];
    v2f b;
    b[0] = sBl1[khalf][ct * 16 + m16];
    b[1] = sBl1[khalf + 1][ct * 16 + m16];
    v8f acc = (v8f){0, 0, 0, 0, 0, 0, 0, 0};
    acc = wmma_f32_4(a, b, acc);
#pragma unroll
    for (int v = 0; v < 8; ++v) t1[rt * 16 + v + hi8][ct * 16 + m16] = acc[v];
  }
  __syncthreads();

  // --- LayerNorm(64) + ReLU per pair-row ---
  if (tid < PP) {
    float vb[64];
    const float4* rp = (const float4*)&t1[tid][0];
    float mean = 0.f;
#pragma unroll
    for (int i = 0; i < 16; ++i) {
      float4 t4 = rp[i];
      vb[4 * i + 0] = t4.x;
      vb[4 * i + 1] = t4.y;
      vb[4 * i + 2] = t4.z;
      vb[4 * i + 3] = t4.w;
      mean += t4.x + t4.y + t4.z + t4.w;
    }
    mean *= (1.f / 64.f);
    float var = 0.f;
#pragma unroll
    for (int c = 0; c < 64; ++c) {
      float d = vb[c] - mean;
      var += d * d;
    }
    float inv = rsqrtf(var * (1.f / 64.f) + 1e-5f);
    float4* wp = (float4*)&t1[tid][0];
#pragma unroll
    for (int i = 0; i < 16; ++i) {
      float4 o;
      o.x = fmaxf((vb[4 * i + 0] - mean) * inv * pb_lnw[4 * i + 0] + pb_lnb[4 * i + 0], 0.f);
      o.y = fmaxf((vb[4 * i + 1] - mean) * inv * pb_lnw[4 * i + 1] + pb_lnb[4 * i + 1], 0.f);
      o.z = fmaxf((vb[4 * i + 2] - mean) * inv * pb_lnw[4 * i + 2] + pb_lnb[4 * i + 2], 0.f);
      o.w = fmaxf((vb[4 * i + 3] - mean) * inv * pb_lnw[4 * i + 3] + pb_lnb[4 * i + 3], 0.f);
      wp[i] = o;
    }
  }
  __syncthreads();

  // --- pb.l2: [64,64] @ [64,64]; 2 tiles per wave, preloaded fragments ---
#pragma unroll
  for (int tt = 0; tt < 2; ++tt) {
    int t = wave + tt * 8;
    int rt = t >> 2, ct = t & 3;
    const int arow = rt * 16 + m16, col = ct * 16 + m16;
    v2f af[16], bf[16];
#pragma unroll
    for (int kk = 0; kk < 16; ++kk)
      af[kk] = *(const v2f*)&t1[arow][kk * 4 + khalf];
#pragma unroll
    for (int kk = 0; kk < 16; ++kk) {
      v2f b;
      b[0] = sW2[kk * 4 + khalf][col];
      b[1] = sW2[kk * 4 + khalf + 1][col];
      bf[kk] = b;
    }
    v8f acc = (v8f){0, 0, 0, 0, 0, 0, 0, 0};
#pragma unroll
    for (int kk = 0; kk < 16; ++kk) acc = wmma_f32_4(af[kk], bf[kk], acc);
#pragma unroll
    for (int v = 0; v < 8; ++v)
      peb[rt * 16 + v + hi8][col] = acc[v] + pb_l2b[col];
  }
  __syncthreads();

  // --- rel = gather(key) - q + peb, into t1 ---
  {
    int p = tid >> 2, c0 = (tid & 3) << 4;
    int ic = idxb[p] < 0 ? 0 : idxb[p];
    int qn = p >> 4;
    const float4* kp = (const float4*)(kbuf + (size_t)ic * 64 + c0);
#pragma unroll
    for (int j = 0; j < 4; ++j) {
      float4 kv = kp[j];
      int c = c0 + j * 4;
      float4 q4 = *(const float4*)&qs[qn][c];
      float4 p4 = *(const float4*)&peb[p][c];
      float4 o;
      o.x = kv.x - q4.x + p4.x;
      o.y = kv.y - q4.y + p4.y;
      o.z = kv.z - q4.z + p4.z;
      o.w = kv.w - q4.w + p4.w;
      *(float4*)&t1[p][c] = o;
    }
  }
  __syncthreads();

  // --- we.l1: [64,64] @ [64,8(->16 zero pad)]; waves 0..3, 1 tile each ---
  if (wave < 4) {
    const int arow = wave * 16 + m16;
    v2f af[16], bf[16];
#pragma unroll
    for (int kk = 0; kk < 16; ++kk)
      af[kk] = *(const v2f*)&t1[arow][kk * 4 + khalf];
#pragma unroll
    for (int kk = 0; kk < 16; ++kk) {
      v2f b;
      b[0] = sWe[kk * 4 + khalf][m16];
      b[1] = sWe[kk * 4 + khalf + 1][m16];
      bf[kk] = b;
    }
    v8f acc = (v8f){0, 0, 0, 0, 0, 0, 0, 0};
#pragma unroll
    for (int kk = 0; kk < 16; ++kk) acc = wmma_f32_4(af[kk], bf[kk], acc);
#pragma unroll
    for (int v = 0; v < 8; ++v) wl[wave * 16 + v + hi8][m16] = acc[v];
  }
  __syncthreads();

  // --- per-pair tail: +bias, LN(8), ReLU, 8x8 linear ---
  if (tid < PP) {
    float h[8];
    float mean = 0.f;
#pragma unroll
    for (int g = 0; g < 8; ++g) {
      h[g] = wl[tid][g] + we_l1b[g];
      mean += h[g];
    }
    mean *= 0.125f;
    float var = 0.f;
#pragma unroll
    for (int g = 0; g < 8; ++g) {
      float d = h[g] - mean;
      var += d * d;
    }
    float inv = rsqrtf(var * 0.125f + 1e-5f);
#pragma unroll
    for (int g = 0; g < 8; ++g)
      h[g] = fmaxf((h[g] - mean) * inv * we_lnw[g] + we_lnb[g], 0.f);
#pragma unroll
    for (int g = 0; g < 8; ++g) {
      float o = we_l2b[g];
#pragma unroll
      for (int j = 0; j < 8; ++j) o += h[j] * we_l2w[j * 8 + g];
      wl[tid][g] = o;
    }
  }
  __syncthreads();

  // --- softmax over K per (query, group), then mask ---
  if (tid < QB * 8) {
    int qn = tid >> 3, g = tid & 7;
    float mx = -3.4e38f;
    for (int k = 0; k < KN; ++k) mx = fmaxf(mx, wl[qn * 16 + k][g]);
    float e[KN];
    float s = 0.f;
    for (int k = 0; k < KN; ++k) {
      e[k] = __expf(wl[qn * 16 + k][g] - mx);
      s += e[k];
    }
    float invs = 1.f / s;
    for (int k = 0; k < KN; ++k) {
      float m = (idxb[qn * 16 + k] >= 0) ? 1.f : 0.f;
      wl[qn * 16 + k][g] = e[k] * invs * m;
    }
  }
  // --- concurrently: vv = gather(value) + peb, into t1 (t1 unused now) ---
  {
    int p = tid >> 2, c0 = (tid & 3) << 4;
    int ic = idxb[p] < 0 ? 0 : idxb[p];
    const float4* vp = (const float4*)(vbuf + (size_t)ic * 64 + c0);
#pragma unroll
    for (int j = 0; j < 4; ++j) {
      float4 vv = vp[j];
      int c = c0 + j * 4;
      float4 p4 = *(const float4*)&peb[p][c];
      float4 o;
      o.x = vv.x + p4.x;
      o.y = vv.y + p4.y;
      o.z = vv.z + p4.z;
      o.w = vv.w + p4.w;
      *(float4*)&t1[p][c] = o;
    }
  }
  __syncthreads();

  // --- grouped weighted sum: out[n][c] = sum_k vv[k][c] * w[n][k][c/8] ---
  {
    int nn = tid >> 6, c = tid & 63, g = c >> 3;
    float acc = 0.f;
#pragma unroll
    for (int k = 0; k < KN; ++k)
      acc += t1[nn * 16 + k][c] * wl[nn * 16 + k][g];
    xout[(size_t)(n0 + nn) * 64 + c] = acc;
  }
}

// ---------------------------------------------------------------------------
// out[row] = (relu?)(idn[row] + LayerNorm(x[row]) * lnw + lnb), C=64.
// idn may alias out (in-place residual) -> no restrict on those.
// ---------------------------------------------------------------------------
__global__ __launch_bounds__(256) void addln_kernel(
    const float* idn, const float* __restrict__ x,
    const float* __restrict__ lnw, const float* __restrict__ lnb, float* out,
    int R, int reluOut) {
  int row = blockIdx.x * 256 + threadIdx.x;
  if (row >= R) return;
  const float4* xr = (const float4*)(x + (size_t)row * 64);
  float v[64];
  float mean = 0.f;
#pragma unroll
  for (int i = 0; i < 16; ++i) {
    float4 t = xr[i];
    v[4 * i + 0] = t.x;
    v[4 * i + 1] = t.y;
    v[4 * i + 2] = t.z;
    v[4 * i + 3] = t.w;
    mean += t.x + t.y + t.z + t.w;
  }
  mean *= (1.f / 64.f);
  float var = 0.f;
#pragma unroll
  for (int c = 0; c < 64; ++c) {
    float d = v[c] - mean;
    var += d * d;
  }
  float inv = rsqrtf(var * (1.f / 64.f) + 1e-5f);
  const float4* ir = (const float4*)(idn + (size_t)row * 64);
  float4* orow = (float4*)(out + (size_t)row * 64);
#pragma unroll
  for (int i = 0; i < 16; ++i) {
    float4 id4 = ir[i];
    float4 o;
    o.x = id4.x + (v[4 * i + 0] - mean) * inv * lnw[4 * i + 0] + lnb[4 * i + 0];
    o.y = id4.y + (v[4 * i + 1] - mean) * inv * lnw[4 * i + 1] + lnb[4 * i + 1];
    o.z = id4.z + (v[4 * i + 2] - mean) * inv * lnw[4 * i + 2] + lnb[4 * i + 2];
    o.w = id4.w + (v[4 * i + 3] - mean) * inv * lnw[4 * i + 3] + lnb[4 * i + 3];
    if (reluOut) {
      o.x = fmaxf(o.x, 0.f);
      o.y = fmaxf(o.y, 0.f);
      o.z = fmaxf(o.z, 0.f);
      o.w = fmaxf(o.w, 0.f);
    }
    orow[i] = o;
  }
}

// ---------------------------------------------------------------------------
struct GvaP {
  const float *k_lb, *k_lw, *k_nb, *k_nw;
  const float *pb_l1b, *pb_l1w, *pb_l2b, *pb_l2w, *pb_lnb, *pb_lnw;
  const float *q_lb, *q_lw, *q_nb, *q_nw;
  const float *v_b, *v_w;
  const float *we_l1b, *we_l1w, *we_l2b, *we_l2w, *we_lnb, *we_lnw;
};

extern "C" void kernel_launch(void* const* d_in, const int* in_sizes, int n_in,
                              void* d_out, int out_size, void* d_ws,
                              size_t ws_size, hipStream_t stream) {
  const float* coord = (const float*)d_in[0];
  const float* feat = (const float*)d_in[1];
  const float* skip_coord = (const float*)d_in[2];
  const float* skip_feat = (const float*)d_in[3];
  const int* idx_sa = (const int*)d_in[4];
  const int* idx_ca = (const int*)d_in[5];

  const int N = in_sizes[1] / 64;
  const int M = in_sizes[3] / 64;

  auto P = [&](int i) { return (const float*)d_in[i]; };
  auto loadGva = [&](int b) {
    GvaP g;
    g.k_lb = P(b + 0);  g.k_lw = P(b + 1);  g.k_nb = P(b + 2);  g.k_nw = P(b + 3);
    g.pb_l1b = P(b + 4); g.pb_l1w = P(b + 5);
    g.pb_l2b = P(b + 6); g.pb_l2w = P(b + 7);
    g.pb_lnb = P(b + 8); g.pb_lnw = P(b + 9);
    g.q_lb = P(b + 10); g.q_lw = P(b + 11); g.q_nb = P(b + 12); g.q_nw = P(b + 13);
    g.v_b = P(b + 14);  g.v_w = P(b + 15);
    g.we_l1b = P(b + 16); g.we_l1w = P(b + 17);
    g.we_l2b = P(b + 18); g.we_l2w = P(b + 19);
    g.we_lnb = P(b + 20); g.we_lnw = P(b + 21);
    return g;
  };
  // jax tree-flatten order (sorted dict keys): ca, mlp, norm1..3, sa
  GvaP ca = loadGva(6);
  const float* mlp_l1b = P(28);
  const float* mlp_l1w = P(29);
  const float* mlp_l2b = P(30);
  const float* mlp_l2w = P(31);
  const float* n1b = P(32);
  const float* n1w = P(33);
  const float* n2b = P(34);
  const float* n2w = P(35);
  const float* n3b = P(36);
  const float* n3w = P(37);
  GvaP sa = loadGva(38);

  float* ws = (float*)d_ws;
  size_t NC = (size_t)N * 64, MC = (size_t)M * 64;
  size_t big = NC > MC ? NC : MC;
  float* qb = ws;  // also reused as MLP hidden
  float* kb = qb + big;
  float* vb = kb + big;
  float* xb = vb + big;
  float* fc = xb + big;
  float* out = (float*)d_out;

  dim3 blk(256);
  auto gl = [](int R) { return dim3((unsigned)((R + 127) / 128)); };
  dim3 gRowN((unsigned)((N + 255) / 256));
  dim3 gGva((unsigned)(N / 4));

  // ---- self-attention block ----
  linear_kernel<1><<<gl(N), blk, 0, stream>>>(feat, sa.q_lw, sa.q_lb, sa.q_nw, sa.q_nb, qb, N);
  linear_kernel<1><<<gl(N), blk, 0, stream>>>(feat, sa.k_lw, sa.k_lb, sa.k_nw, sa.k_nb, kb, N);
  linear_kernel<0><<<gl(N), blk, 0, stream>>>(feat, sa.v_w, sa.v_b, nullptr, nullptr, vb, N);
  gva_kernel<<<gGva, blk, 0, stream>>>(qb, kb, vb, coord, coord, idx_sa,
                                       sa.pb_l1w, sa.pb_l1b, sa.pb_lnw, sa.pb_lnb,
                                       sa.pb_l2w, sa.pb_l2b, sa.we_l1w, sa.we_l1b,
                                       sa.we_lnw, sa.we_lnb, sa.we_l2w, sa.we_l2b, xb);
  addln_kernel<<<gRowN, blk, 0, stream>>>(feat, xb, n1w, n1b, fc, N, 0);

  // ---- cross-attention block ----
  linear_kernel<1><<<gl(N), blk, 0, stream>>>(fc, ca.q_lw, ca.q_lb, ca.q_nw, ca.q_nb, qb, N);
  linear_kernel<1><<<gl(M), blk, 0, stream>>>(skip_feat, ca.k_lw, ca.k_lb, ca.k_nw, ca.k_nb, kb, M);
  linear_kernel<0><<<gl(M), blk, 0, stream>>>(skip_feat, ca.v_w, ca.v_b, nullptr, nullptr, vb, M);
  gva_kernel<<<gGva, blk, 0, stream>>>(qb, kb, vb, skip_coord, coord, idx_ca,
                                       ca.pb_l1w, ca.pb_l1b, ca.pb_lnw, ca.pb_lnb,
                                       ca.pb_l2w, ca.pb_l2b, ca.we_l1w, ca.we_l1b,
                                       ca.we_lnw, ca.we_lnb, ca.we_l2w, ca.we_l2b, xb);
  addln_kernel<<<gRowN, blk, 0, stream>>>(fc, xb, n2w, n2b, fc, N, 0);

  // ---- MLP block ----
  linear_kernel<2><<<gl(N), blk, 0, stream>>>(fc, mlp_l1w, mlp_l1b, nullptr, nullptr, qb, N);
  linear_kernel<0><<<gl(N), blk, 0, stream>>>(qb, mlp_l2w, mlp_l2b, nullptr, nullptr, xb, N);
  addln_kernel<<<gRowN, blk, 0, stream>>>(fc, xb, n3w, n3b, out, N, 1);
}